// Where2commFusion_41661182771836
// MI455X (gfx1250) — compile-verified
//
#include <hip/hip_runtime.h>

// Problem constants (from reference): B=4, N=5, C=64, H=100, W=252
#define BB 4
#define NN 5
#define CC 64
#define HH 100
#define WW 252
#define HWSZ (HH * WW)             // 25200
#define PIX (BB * HH * WW)         // 100800 output pixels
#define CPT 16                     // channels per thread
#define NGRP (CC / CPT)            // 4 channel groups
#define TOTAL_THREADS (PIX * NGRP) // 403200 (exact multiple of 256)
#define BLOCK 256
#define OUT_ELEMS (BB * CC * HWSZ) // 6451200 (scalar comm-rate follows at this index)

// ---- CDNA5 async gather-to-LDS path (guarded so the file always compiles) ----
#if defined(__has_builtin)
#if __has_builtin(__builtin_amdgcn_global_load_async_to_lds_b32)
#define HAVE_ASYNC_LDS 1
#endif
#endif
#ifndef HAVE_ASYNC_LDS
#define HAVE_ASYNC_LDS 0
#endif

#if HAVE_ASYNC_LDS
#define STR_(x) #x
#define STR(x) STR_(x)
#if __has_builtin(__builtin_amdgcn_s_wait_asynccnt)
#define WAIT_ASYNC(n) __builtin_amdgcn_s_wait_asynccnt(n)
#else
#define WAIT_ASYNC(n) asm volatile("s_wait_asynccnt " STR(n) ::: "memory")
#endif
// Signature confirmed by hipcc diagnostic: (global int* src, local int* dst, imm offset, imm cpol)
typedef __attribute__((address_space(1))) int* gas1_ptr;
typedef __attribute__((address_space(3))) int* lds3_ptr;
#endif

__global__ __launch_bounds__(BLOCK) void where2comm_fuse_kernel(
    const float* __restrict__ x,    // (B*N, C, H, W)
    const float* __restrict__ ptm,  // (B, N, N, 2, 3)
    float* __restrict__ out)        // (B, C, H, W) + 1 scalar
{
  const int tid = blockIdx.x * BLOCK + threadIdx.x;
  const int g  = tid / PIX;        // channel group 0..3 (uniform per wave)
  const int p  = tid - g * PIX;    // pixel id
  const int b  = p / HWSZ;
  const int hw = p - b * HWSZ;
  const int h  = hw / WW;
  const int w  = hw - h * WW;

  // Normalized grid coords (align_corners=False style, matches reference)
  const float gx = (2.0f * (float)w + 1.0f) / (float)WW - 1.0f;
  const float gy = (2.0f * (float)h + 1.0f) / (float)HH - 1.0f;

  // Per agent: two row base pointers (columns xs, xs+1 read as a pair) and
  // four weights with clamp/validity selection pre-folded in.
  const float* rowp[NN][2];
  float        wt[NN][4];   // [row0.x, row0.y, row1.x, row1.y]

#pragma unroll
  for (int n = 0; n < NN; ++n) {
    // theta = pairwise_t_matrix[b, 0, n, :, :]
    const float* t = ptm + (size_t)(((b * NN + 0) * NN + n) * 6);
    const float cx = t[0] * gx + t[1] * gy + t[2];
    const float cy = t[3] * gx + t[4] * gy + t[5];
    const float ix = ((cx + 1.0f) * (float)WW - 1.0f) * 0.5f;
    const float iy = ((cy + 1.0f) * (float)HH - 1.0f) * 0.5f;
    const float x0f = floorf(ix), y0f = floorf(iy);
    const float x1f = x0f + 1.0f, y1f = y0f + 1.0f;
    const float wx1 = ix - x0f, wx0 = 1.0f - wx1;
    const float wy1 = iy - y0f, wy0 = 1.0f - wy1;
    const float vx0 = (x0f >= 0.0f && x0f <= (float)(WW - 1)) ? 1.0f : 0.0f;
    const float vx1 = (x1f >= 0.0f && x1f <= (float)(WW - 1)) ? 1.0f : 0.0f;
    const float vy0 = (y0f >= 0.0f && y0f <= (float)(HH - 1)) ? 1.0f : 0.0f;
    const float vy1 = (y1f >= 0.0f && y1f <= (float)(HH - 1)) ? 1.0f : 0.0f;
    const int xc0 = min(max((int)x0f, 0), WW - 1);
    const int xc1 = min(max((int)x1f, 0), WW - 1);
    const int yc0 = min(max((int)y0f, 0), HH - 1);
    const int yc1 = min(max((int)y1f, 0), HH - 1);

    // Both clamped columns provably lie in {xs, xs+1}; in-bounds pair base.
    const int xs = min(xc0, WW - 2);
    const float ax = wx0 * vx0;          // weight attached to column xc0
    const float bx = wx1 * vx1;          // weight attached to column xc1
    const float wxA = ((xc0 == xs) ? ax : 0.0f) + ((xc1 == xs) ? bx : 0.0f);
    const float wxB = (ax + bx) - wxA;
    const float wy0v = wy0 * vy0;
    const float wy1v = wy1 * vy1;
    wt[n][0] = wxA * wy0v;
    wt[n][1] = wxB * wy0v;
    wt[n][2] = wxA * wy1v;
    wt[n][3] = wxB * wy1v;

    const int base = ((b * NN + n) * CC + g * CPT) * HWSZ;
    rowp[n][0] = x + base + yc0 * WW + xs;
    rowp[n][1] = x + base + yc1 * WW + xs;
  }

  const size_t out0 = (size_t)(b * CC + g * CPT) * HWSZ + (size_t)hw;

#if HAVE_ASYNC_LDS
  // Double-buffered async gather: 2 bufs x 256 threads x 5 agents x 2 rows (float2 pair).
  __shared__ float2 stage[2][BLOCK][NN * 2];  // 40 KB of 320 KB LDS
  const int tx = threadIdx.x;

  // Each pair is fetched with two b32 async ops sharing one address VGPR pair:
  // INST_OFFSET is added to BOTH the global and the LDS address, and the pair
  // components are +4B apart in both spaces, so offset 0 / 4 reuses the address.
#define ISSUE_BATCH(CH, BUF)                                                   \
  _Pragma("unroll") for (int n = 0; n < NN; ++n) {                             \
    const float* gA = rowp[n][0] + (CH) * HWSZ;                                \
    const float* gB = rowp[n][1] + (CH) * HWSZ;                                \
    lds3_ptr lA = (lds3_ptr)&stage[(BUF)][tx][2 * n + 0];                      \
    lds3_ptr lB = (lds3_ptr)&stage[(BUF)][tx][2 * n + 1];                      \
    __builtin_amdgcn_global_load_async_to_lds_b32((gas1_ptr)gA, lA, 0, 0);     \
    __builtin_amdgcn_global_load_async_to_lds_b32((gas1_ptr)gA, lA, 4, 0);     \
    __builtin_amdgcn_global_load_async_to_lds_b32((gas1_ptr)gB, lB, 0, 0);     \
    __builtin_amdgcn_global_load_async_to_lds_b32((gas1_ptr)gB, lB, 4, 0);     \
  }

  ISSUE_BATCH(0, 0)  // prologue: channel 0 into buffer 0

  for (int c = 0; c < CPT; ++c) {
    const int buf = c & 1;
    if (c + 1 < CPT) {
      ISSUE_BATCH(c + 1, buf ^ 1)
      WAIT_ASYNC(20);   // <=20 outstanding -> batch c complete (in-order)
    } else {
      WAIT_ASYNC(0);
    }
    const float2* st = stage[buf][tx];
    float acc = -INFINITY;
#pragma unroll
    for (int n = 0; n < NN; ++n) {
      const float2 f0 = st[2 * n + 0];
      const float2 f1 = st[2 * n + 1];
      const float v = f0.x * wt[n][0] + f0.y * wt[n][1]
                    + f1.x * wt[n][2] + f1.y * wt[n][3];
      acc = fmaxf(acc, v);
    }
    out[out0 + (size_t)c * HWSZ] = acc;
  }
#undef ISSUE_BATCH
#else
  // Fallback: direct global gathers with prefetch (global_prefetch_b8).
#pragma unroll 4
  for (int c = 0; c < CPT; ++c) {
    float acc = -INFINITY;
#pragma unroll
    for (int n = 0; n < NN; ++n) {
      const float* gA = rowp[n][0] + c * HWSZ;
      const float* gB = rowp[n][1] + c * HWSZ;
      if (c + 2 < CPT) __builtin_prefetch(gA + 2 * HWSZ, 0, 0);
      const float v = gA[0] * wt[n][0] + gA[1] * wt[n][1]
                    + gB[0] * wt[n][2] + gB[1] * wt[n][3];
      acc = fmaxf(acc, v);
    }
    out[out0 + (size_t)c * HWSZ] = acc;
  }
#endif

  // communication_rates scalar (second tuple element) = 0.0
  if (tid == 0) out[OUT_ELEMS] = 0.0f;
}

extern "C" void kernel_launch(void* const* d_in, const int* in_sizes, int n_in,
                              void* d_out, int out_size, void* d_ws, size_t ws_size,
                              hipStream_t stream) {
  (void)in_sizes; (void)n_in; (void)d_ws; (void)ws_size; (void)out_size;
  const float* x   = (const float*)d_in[0];  // (B*N, C, H, W)
  const float* ptm = (const float*)d_in[3];  // (B, N, N, 2, 3); d_in[1]=rm, d_in[2]=record_len unused
  float* out = (float*)d_out;

  dim3 grid(TOTAL_THREADS / BLOCK);  // 1575 blocks
  dim3 block(BLOCK);                 // 8 wave32s
  where2comm_fuse_kernel<<<grid, block, 0, stream>>>(x, ptm, out);
}